// Net_6743098654994
// MI455X (gfx1250) — compile-verified
//
#include <hip/hip_runtime.h>
#include <math.h>

typedef __attribute__((ext_vector_type(2))) float v2f;
typedef __attribute__((ext_vector_type(4))) float v4f;
typedef __attribute__((ext_vector_type(8))) float v8f;

#define H_DIM 256
#define IO_DIM 32
#define ACT_EPS 0.01f

// ---------------------------------------------------------------------------
// Kernel 1: fold Wp into Wx_h:  Wc = Wx[:, :H] @ Wp  (256x32),
//           bvec = Wx[:, :H] @ bp + b,  wxflag = Wx[:, H]
// ---------------------------------------------------------------------------
__global__ void setup_kernel(const float* __restrict__ Wx, const float* __restrict__ Wp,
                             const float* __restrict__ bp, const float* __restrict__ b,
                             float* __restrict__ Wc, float* __restrict__ bvec,
                             float* __restrict__ wxflag) {
    int tid = blockIdx.x * blockDim.x + threadIdx.x;   // 0..8191
    int i = tid >> 5;                                  // hidden row
    int k = tid & 31;                                  // io col
    float acc = 0.f;
    for (int j = 0; j < H_DIM; ++j)
        acc += Wx[i * (H_DIM + 1) + j] * Wp[j * IO_DIM + k];
    Wc[i * IO_DIM + k] = acc;
    if (k == 0) {
        float bb = b[i];
        for (int j = 0; j < H_DIM; ++j)
            bb += Wx[i * (H_DIM + 1) + j] * bp[j];
        bvec[i] = bb;
        wxflag[i] = Wx[i * (H_DIM + 1) + H_DIM];
    }
}

// ---------------------------------------------------------------------------
// Kernel 2: U[T,256] = X[T,32] @ Wc^T + bvec   via V_WMMA_F32_16X16X4_F32
// One wave per 16x16 output tile; K=32 -> 8 WMMAs of K=4.
// f32 A 16x4 layout: lane l: M=l&15, VGPR0=A[M][2h], VGPR1=A[M][2h+1], h=l>>4
// f32 B 4x16 layout: lane l: N=l&15, VGPR0=B[2h][N], VGPR1=B[2h+1][N]
// f32 D 16x16:       lane l: VGPRv = D[v + 8h][l&15]
// ---------------------------------------------------------------------------
__global__ void u_gemm_kernel(const float* __restrict__ X, const float* __restrict__ Wc,
                              const float* __restrict__ bvec, float* __restrict__ U,
                              int Mtiles) {
    int wave  = (blockIdx.x * blockDim.x + threadIdx.x) >> 5;
    int lane  = threadIdx.x & 31;
    int mtile = wave >> 4;          // 16 n-tiles
    int ntile = wave & 15;
    if (mtile >= Mtiles) return;    // wave-uniform guard (EXEC all-1s for WMMA)
    int half = lane >> 4, l15 = lane & 15;
    int arow = mtile * 16 + l15;
    int brow = ntile * 16 + l15;
    v8f c = {};
#pragma unroll
    for (int kc = 0; kc < 8; ++kc) {
        int kk = kc * 4 + 2 * half;
        v2f a  = *(const v2f*)(X  + arow * IO_DIM + kk);
        v2f bf = *(const v2f*)(Wc + brow * IO_DIM + kk);
        c = __builtin_amdgcn_wmma_f32_16x16x4_f32(false, a, false, bf, (short)0, c,
                                                  false, false);
    }
    float bv = bvec[ntile * 16 + l15];
#pragma unroll
    for (int v = 0; v < 8; ++v) {
        int M = v + 8 * half;
        U[(mtile * 16 + M) * H_DIM + ntile * 16 + l15] = c[v] + bv;
    }
}

// ---------------------------------------------------------------------------
// Kernel 3: the sequential ACT scan. Single workgroup (256 thr = 8 waves),
// Ws (256KB fp32) resident in LDS, K-interleaved for conflict-free b128 reads.
// Thread i owns hidden row i. One barrier per ponder step.
// ---------------------------------------------------------------------------
__global__ void scan_kernel(const float* __restrict__ Ws, const float* __restrict__ U,
                            const float* __restrict__ h0, const float* __restrict__ wxflag,
                            const float* __restrict__ wh, const float* __restrict__ bh_p,
                            float* __restrict__ S, float* __restrict__ pc_out, int T) {
    extern __shared__ float smem[];
    float* Wmat = smem;                    // 64*256*4 = 65536 floats (256 KB)
    float* sbuf = smem + 65536;            // 2 * 256 (double-buffered state)
    float* wred = smem + 65536 + 512;      // 2 * 8   (double-buffered wave sums)

    int i    = threadIdx.x;                // 0..255
    int lane = i & 31, wv = i >> 5;

    // Stage Ws row i into interleaved LDS: Wmat[k4][i][c] = Ws[i][4*k4+c]
#pragma unroll 4
    for (int k4 = 0; k4 < 64; ++k4) {
        v4f w = *(const v4f*)(Ws + i * H_DIM + k4 * 4);
        *(v4f*)(Wmat + k4 * 1024 + i * 4) = w;
    }
    sbuf[i] = h0[i];

    float u_next = U[i];
    float whi = wh[i];
    float bhv = bh_p[0];
    float wfl = wxflag[i];
    float pc = 0.f;
    float Tf = (float)T;
    int cur = 0;
    __syncthreads();

    for (int t = 0; t < T; ++t) {
        float u = u_next;
        if (t + 1 < T) u_next = U[(t + 1) * H_DIM + i];   // pipelined prefetch

        float cum = 0.f, accv = 0.f, n_up = 0.f, rem = 0.f;
        bool still = true;
#pragma unroll
        for (int n = 0; n < 4; ++n) {
            // dot = Ws[i,:] . s  (4 accumulators to break the FMA chain)
            float a0 = 0.f, a1 = 0.f, a2 = 0.f, a3 = 0.f;
            const float* sb = sbuf + cur * H_DIM;
#pragma unroll 8
            for (int k4 = 0; k4 < 64; ++k4) {
                v4f w  = *(const v4f*)(Wmat + k4 * 1024 + i * 4);
                v4f sv = *(const v4f*)(sb + k4 * 4);      // broadcast read
                a0 += w.x * sv.x; a1 += w.y * sv.y;
                a2 += w.z * sv.z; a3 += w.w * sv.w;
            }
            float pre = u + (n == 0 ? wfl : 0.f) + (a0 + a1) + (a2 + a3);
            float s_new = tanhf(pre);

            // halting unit: p = sigmoid(wh . s + bh) via wave + LDS reduction
            float part = whi * s_new;
#pragma unroll
            for (int m = 16; m >= 1; m >>= 1) part += __shfl_xor(part, m, 32);
            int nxt = cur ^ 1;
            if (lane == 0) wred[nxt * 8 + wv] = part;
            sbuf[nxt * H_DIM + i] = s_new;
            __syncthreads();

            float hsum = 0.f;
#pragma unroll
            for (int q = 0; q < 8; ++q) hsum += wred[nxt * 8 + q];
            float p = 1.0f / (1.0f + expf(-(hsum + bhv)));

            bool halt = (cum + p > 1.0f - ACT_EPS) || (n == 3);
            float w = still ? (halt ? (1.0f - cum) : p) : 0.f;
            accv += w * s_new;
            rem  += (still && halt) ? (1.0f - cum) : 0.f;
            n_up += still ? 1.f : 0.f;
            cum  += w;
            still = still && !halt;
            cur = nxt;
        }
        // acc_s becomes next h; also emit it for the output GEMM
        int nxt = cur ^ 1;
        sbuf[nxt * H_DIM + i] = accv;
        S[t * H_DIM + i] = accv;
        __syncthreads();
        cur = nxt;
        pc = (pc + (n_up + rem)) / Tf;   // true division, matches reference
    }
    if (i == 0) pc_out[0] = pc;
}

// ---------------------------------------------------------------------------
// Kernel 4: ys[T,32] = S[T,256] @ Wo^T + bo  via V_WMMA_F32_16X16X4_F32
// ---------------------------------------------------------------------------
__global__ void y_gemm_kernel(const float* __restrict__ S, const float* __restrict__ Wo,
                              const float* __restrict__ bo, float* __restrict__ Y,
                              int Mtiles) {
    int wave  = (blockIdx.x * blockDim.x + threadIdx.x) >> 5;
    int lane  = threadIdx.x & 31;
    int mtile = wave >> 1;          // 2 n-tiles
    int ntile = wave & 1;
    if (mtile >= Mtiles) return;
    int half = lane >> 4, l15 = lane & 15;
    int arow = mtile * 16 + l15;
    int brow = ntile * 16 + l15;
    v8f c = {};
#pragma unroll 8
    for (int kc = 0; kc < 64; ++kc) {
        int kk = kc * 4 + 2 * half;
        v2f a  = *(const v2f*)(S  + arow * H_DIM + kk);
        v2f bf = *(const v2f*)(Wo + brow * H_DIM + kk);
        c = __builtin_amdgcn_wmma_f32_16x16x4_f32(false, a, false, bf, (short)0, c,
                                                  false, false);
    }
    float bv = bo[ntile * 16 + l15];
#pragma unroll
    for (int v = 0; v < 8; ++v) {
        int M = v + 8 * half;
        Y[(mtile * 16 + M) * IO_DIM + ntile * 16 + l15] = c[v] + bv;
    }
}

// ---------------------------------------------------------------------------
extern "C" void kernel_launch(void* const* d_in, const int* in_sizes, int n_in,
                              void* d_out, int out_size, void* d_ws, size_t ws_size,
                              hipStream_t stream) {
    (void)n_in; (void)out_size; (void)ws_size;
    const float* x  = (const float*)d_in[0];
    const float* h0 = (const float*)d_in[1];
    const float* Wp = (const float*)d_in[2];
    const float* bp = (const float*)d_in[3];
    const float* Wx = (const float*)d_in[4];
    const float* Ws = (const float*)d_in[5];
    const float* b  = (const float*)d_in[6];
    const float* wh = (const float*)d_in[7];
    const float* bh = (const float*)d_in[8];
    const float* Wo = (const float*)d_in[9];
    const float* bo = (const float*)d_in[10];

    int T = in_sizes[0] / IO_DIM;          // 8192

    float* ws     = (float*)d_ws;
    float* Wc     = ws;                    // 256*32
    float* bvec   = ws + 8192;             // 256
    float* wxflag = ws + 8448;             // 256
    float* U      = ws + 8704;             // T*256
    float* S      = U + (size_t)T * H_DIM; // T*256

    float* Y  = (float*)d_out;             // T*32
    float* pc = Y + (size_t)T * IO_DIM;    // 1

    size_t shmem = (65536 + 512 + 16) * sizeof(float);  // ~258 KB of the 320 KB WGP LDS
    // Raise the dynamic-LDS cap for the scan kernel (not a stream op -> capture safe).
    (void)hipFuncSetAttribute((const void*)scan_kernel,
                              hipFuncAttributeMaxDynamicSharedMemorySize, (int)shmem);

    setup_kernel<<<32, 256, 0, stream>>>(Wx, Wp, bp, b, Wc, bvec, wxflag);

    int mtiles = T / 16;                   // 512
    u_gemm_kernel<<<(mtiles * 16) / 8, 256, 0, stream>>>(x, Wc, bvec, U, mtiles);

    scan_kernel<<<1, 256, shmem, stream>>>(Ws, U, h0, wxflag, wh, bh, S, pc, T);

    y_gemm_kernel<<<(mtiles * 2) / 8, 256, 0, stream>>>(S, Wo, bo, Y, mtiles);
}